// Capsule_fc_86457691668917
// MI455X (gfx1250) — compile-verified
//
#include <hip/hip_runtime.h>
#include <hip/hip_bf16.h>

// Problem dims (compile-time constants from reference)
#define BATCH 64
#define NIN   256
#define DIN   8
#define NOUT  32
#define DOUT  16
#define RITER 3
#define KDIM  (NIN * DIN)            // 2048
#define NDIM  (NIN * NOUT * DOUT)    // 131072

typedef __attribute__((ext_vector_type(2))) float v2f;
typedef __attribute__((ext_vector_type(8))) float v8f;

// ---------------------------------------------------------------------------
// Projection GEMM: u[b][n] = sum_k x[b][k] * W[n][k] + bias[n]
// One wave computes a 16(M) x 64(N) tile via 4 f32 WMMA accumulators.
// HBM-bound on W (1.07 GB read once); fp32 WMMA keeps precision at no cost.
// ---------------------------------------------------------------------------
__global__ __launch_bounds__(256) void proj_gemm(const float* __restrict__ x,
                                                 const float* __restrict__ W,
                                                 const float* __restrict__ bias,
                                                 float* __restrict__ u) {
    const int lane = threadIdx.x & 31;
    const int wave = threadIdx.x >> 5;
    const int tile = blockIdx.x * 8 + wave;   // 8192 tiles total
    const int nTile = tile & 2047;            // 2048 n-tiles of 64
    const int mTile = tile >> 11;             // 4 m-tiles of 16
    const int m0 = mTile * 16;
    const int n0 = nTile * 64;
    const int half = lane >> 4;               // 0: K=0,1   1: K=2,3
    const int l16  = lane & 15;

    // A fragment (16x4 f32): lane l16 = row M, half selects K pair
    const float* aPtr = x + (long)(m0 + l16) * KDIM + 2 * half;
    // B fragment (4x16 f32): lane l16 = col N, half selects K pair
    const float* bPtr = W + (long)(n0 + l16) * KDIM + 2 * half;

    v8f acc0 = {}, acc1 = {}, acc2 = {}, acc3 = {};

    #pragma unroll 4
    for (int k = 0; k < KDIM; k += 4) {
        v2f a  = *(const v2f*)(aPtr + k);
        v2f b0 = *(const v2f*)(bPtr + k);
        v2f b1 = *(const v2f*)(bPtr + 16L * KDIM + k);
        v2f b2 = *(const v2f*)(bPtr + 32L * KDIM + k);
        v2f b3 = *(const v2f*)(bPtr + 48L * KDIM + k);
        acc0 = __builtin_amdgcn_wmma_f32_16x16x4_f32(false, a, false, b0, (short)0, acc0, false, false);
        acc1 = __builtin_amdgcn_wmma_f32_16x16x4_f32(false, a, false, b1, (short)0, acc1, false, false);
        acc2 = __builtin_amdgcn_wmma_f32_16x16x4_f32(false, a, false, b2, (short)0, acc2, false, false);
        acc3 = __builtin_amdgcn_wmma_f32_16x16x4_f32(false, a, false, b3, (short)0, acc3, false, false);
    }

    // Epilogue: C/D layout -> VGPR v holds row M = v + 8*half, col = l16
    const int col0 = n0 + l16;
    const float bw0 = bias[col0];
    const float bw1 = bias[col0 + 16];
    const float bw2 = bias[col0 + 32];
    const float bw3 = bias[col0 + 48];
    #pragma unroll
    for (int v = 0; v < 8; ++v) {
        const long row = m0 + v + 8 * half;
        float* up = u + row * (long)NDIM + col0;
        up[0]  = acc0[v] + bw0;
        up[16] = acc1[v] + bw1;
        up[32] = acc2[v] + bw2;
        up[48] = acc3[v] + bw3;
    }
}

// ---------------------------------------------------------------------------
// Routing helpers (u is L2-resident: 33.5 MB << 192 MB)
// ---------------------------------------------------------------------------
__global__ __launch_bounds__(256) void init_b(float* __restrict__ blog) {
    int t = blockIdx.x * 256 + threadIdx.x;
    if (t < NIN * NOUT) blog[t] = 0.0f;
}

// c[i][o] = softmax over o of b[i][o]; one thread per input capsule row
__global__ __launch_bounds__(256) void route_softmax(const float* __restrict__ blog,
                                                     float* __restrict__ c) {
    const int i = threadIdx.x;   // 0..255
    float v[NOUT];
    float mx = -3.4e38f;
    #pragma unroll
    for (int o = 0; o < NOUT; ++o) { v[o] = blog[i * NOUT + o]; mx = fmaxf(mx, v[o]); }
    float sum = 0.0f;
    #pragma unroll
    for (int o = 0; o < NOUT; ++o) { v[o] = __expf(v[o] - mx); sum += v[o]; }
    const float inv = 1.0f / sum;
    #pragma unroll
    for (int o = 0; o < NOUT; ++o) c[i * NOUT + o] = v[o] * inv;
}

// s[b][o][:] = squash( sum_i c[i][o] * u[b][i][o][:] )
// One block per (b,o); 256 threads = one per input capsule i; LDS tree reduce.
__global__ __launch_bounds__(256) void route_s(const float* __restrict__ u,
                                               const float* __restrict__ c,
                                               float* __restrict__ sOut) {
    __shared__ float red[256][DOUT + 1];   // +1 pad vs bank conflicts
    const int bo = blockIdx.x;             // 0..2047
    const int bb = bo >> 5;                // batch
    const int o  = bo & 31;                // output capsule
    const int i  = threadIdx.x;            // input capsule

    const float cv = c[i * NOUT + o];
    const float* up = u + ((long)((bb * NIN + i) * NOUT + o)) * DOUT;
    float4 p0 = *(const float4*)(up);
    float4 p1 = *(const float4*)(up + 4);
    float4 p2 = *(const float4*)(up + 8);
    float4 p3 = *(const float4*)(up + 12);
    float vals[DOUT];
    *(float4*)&vals[0]  = p0; *(float4*)&vals[4]  = p1;
    *(float4*)&vals[8]  = p2; *(float4*)&vals[12] = p3;
    #pragma unroll
    for (int d = 0; d < DOUT; ++d) red[i][d] = cv * vals[d];

    for (int off = 128; off > 0; off >>= 1) {
        __syncthreads();
        if (i < off) {
            #pragma unroll
            for (int d = 0; d < DOUT; ++d) red[i][d] += red[i + off][d];
        }
    }
    __syncthreads();

    if (i < DOUT) {   // lanes 0..15 of wave 0
        const float sv = red[0][i];
        float sq = sv * sv;
        #pragma unroll
        for (int m = 8; m > 0; m >>= 1) sq += __shfl_xor(sq, m, 32);
        const float l2 = sqrtf(sq);
        const float f  = l2 / (1.0f + sq);
        sOut[bo * DOUT + i] = sv * f;
    }
}

__device__ __forceinline__ float dot4(float4 a, float4 b) {
    return a.x * b.x + a.y * b.y + a.z * b.z + a.w * b.w;
}

// b[i][o] += sum_b sum_d u[b][i][o][d] * s[b][o][d]
// One wave per (i,o); lanes cover the 64 batches (2 each); shuffle reduce.
__global__ __launch_bounds__(256) void route_agree(const float* __restrict__ u,
                                                   const float* __restrict__ s,
                                                   float* __restrict__ blog) {
    const int w    = (blockIdx.x * 256 + threadIdx.x) >> 5;  // 0..8191
    const int lane = threadIdx.x & 31;
    const int i = w >> 5;   // 0..255
    const int o = w & 31;   // 0..31
    float acc = 0.0f;
    #pragma unroll
    for (int bb = lane; bb < BATCH; bb += 32) {
        const float* up = u + ((long)((bb * NIN + i) * NOUT + o)) * DOUT;
        const float* sp = s + (long)(bb * NOUT + o) * DOUT;
        acc += dot4(*(const float4*)(up),      *(const float4*)(sp));
        acc += dot4(*(const float4*)(up + 4),  *(const float4*)(sp + 4));
        acc += dot4(*(const float4*)(up + 8),  *(const float4*)(sp + 8));
        acc += dot4(*(const float4*)(up + 12), *(const float4*)(sp + 12));
    }
    #pragma unroll
    for (int m = 16; m > 0; m >>= 1) acc += __shfl_xor(acc, m, 32);
    if (lane == 0) blog[i * NOUT + o] += acc;
}

// ---------------------------------------------------------------------------
// Launch sequence (graph-capturable: only kernel launches on `stream`)
// ---------------------------------------------------------------------------
extern "C" void kernel_launch(void* const* d_in, const int* in_sizes, int n_in,
                              void* d_out, int out_size, void* d_ws, size_t ws_size,
                              hipStream_t stream) {
    const float* x   = (const float*)d_in[0];   // (64, 256, 8)
    const float* W   = (const float*)d_in[1];   // (131072, 2048)
    const float* bW  = (const float*)d_in[2];   // (131072,)
    float* out = (float*)d_out;                 // (64, 32, 16)

    char* ws = (char*)d_ws;
    float* u    = (float*)ws;                                       // 64*131072 f32 = 33.55 MB
    float* blog = (float*)(ws + (size_t)BATCH * NDIM * 4);          // 8192 f32
    float* c    = (float*)(ws + (size_t)BATCH * NDIM * 4 + 32768);  // 8192 f32
    float* s    = (float*)(ws + (size_t)BATCH * NDIM * 4 + 65536);  // 32768 f32

    proj_gemm<<<1024, 256, 0, stream>>>(x, W, bW, u);
    init_b<<<(NIN * NOUT + 255) / 256, 256, 0, stream>>>(blog);

    for (int r = 0; r <= RITER; ++r) {
        route_softmax<<<1, 256, 0, stream>>>(blog, c);
        route_s<<<BATCH * NOUT, 256, 0, stream>>>(u, c, (r == RITER) ? out : s);
        if (r < RITER) route_agree<<<1024, 256, 0, stream>>>(u, s, blog);
    }
}